// Attn_65936337928391
// MI455X (gfx1250) — compile-verified
//
#include <hip/hip_runtime.h>
#include <math.h>

typedef __attribute__((ext_vector_type(2))) float v2f;
typedef __attribute__((ext_vector_type(4))) float v4f;
typedef __attribute__((ext_vector_type(8))) float v8f;

#define S_LEN 2048
#define B_LEN 32
#define H_LEN 1024

// ---------------------------------------------------------------------------
// Kernel 1: v[b,h] = sum_k hidden[b,k] * W[k,h]   (M=32, N=1024, K=1024)
// Uses V_WMMA_F32_16X16X4_F32 (exact fp32). One wave per 16x16 output tile.
// A (16x4):  lane 0-15 -> M=lane, VGPRs hold K = 2*half + {0,1}
// B (4x16):  lane 0-15 -> N=lane, VGPRs hold K = 2*half + {0,1}   (half=lane>>4)
// C/D (16x16): VGPR r -> M = r + 8*half, N = lane&15
// ---------------------------------------------------------------------------
__global__ __launch_bounds__(32)
void proj_wmma(const float* __restrict__ hidden,
               const float* __restrict__ W,
               float* __restrict__ v_out) {
    const int lane = threadIdx.x;          // 0..31, all lanes active (EXEC = all 1s)
    const int n0   = blockIdx.x * 16;      // h-tile
    const int m0   = blockIdx.y * 16;      // b-tile
    const int l15  = lane & 15;
    const int half = lane >> 4;

    const float* aRow = hidden + (size_t)(m0 + l15) * H_LEN + 2 * half; // +kk
    const float* bCol = W + (size_t)(2 * half) * H_LEN + n0 + l15;      // +kk*H

    v8f acc = {};
    #pragma unroll 4
    for (int kk = 0; kk < H_LEN; kk += 4) {
        v2f a, b;
        a.x = aRow[kk];
        a.y = aRow[kk + 1];
        b.x = bCol[(size_t)kk * H_LEN];
        b.y = bCol[(size_t)(kk + 1) * H_LEN];
        // D = A*B + C : v_wmma_f32_16x16x4_f32
        acc = __builtin_amdgcn_wmma_f32_16x16x4_f32(
            /*neg_a=*/false, a, /*neg_b=*/false, b,
            /*c_mod=*/(short)0, acc, /*reuse_a=*/false, /*reuse_b=*/false);
    }

    #pragma unroll
    for (int r = 0; r < 8; ++r) {
        const int m = m0 + r + 8 * half;
        v_out[(size_t)m * H_LEN + n0 + l15] = acc[r];
    }
}

// ---------------------------------------------------------------------------
// Kernel 2: attn[b,s] = v[b,:] . enc[s,b,:]
// Bandwidth-bound: streams 256 MB of enc exactly once (non-temporal), v[b,:]
// cached in LDS per block. Block = 256 threads = 8 waves; wave w -> s-index.
// ---------------------------------------------------------------------------
__global__ __launch_bounds__(256)
void energies_kernel(const float* __restrict__ enc,
                     const float* __restrict__ v,
                     float* __restrict__ attn) {
    __shared__ float lds_v[H_LEN];
    const int b = blockIdx.y;
    const int t = threadIdx.x;

    #pragma unroll
    for (int j = 0; j < 4; ++j)
        lds_v[t + j * 256] = v[(size_t)b * H_LEN + t + j * 256];
    __syncthreads();

    const int wave = t >> 5;
    const int lane = t & 31;
    const int s    = blockIdx.x * 8 + wave;

    const v4f* ep = reinterpret_cast<const v4f*>(
        enc + (size_t)s * (B_LEN * H_LEN) + (size_t)b * H_LEN);
    const v4f* vp = reinterpret_cast<const v4f*>(lds_v);

    float partial = 0.f;
    #pragma unroll
    for (int i = 0; i < 8; ++i) {
        const int q = i * 32 + lane;             // which float4 of the 256
        v4f e  = __builtin_nontemporal_load(&ep[q]);  // streaming: TH=NT
        v4f vv = vp[q];
        partial += e.x * vv.x + e.y * vv.y + e.z * vv.z + e.w * vv.w;
    }
    // wave32 butterfly reduction
    #pragma unroll
    for (int off = 16; off > 0; off >>= 1)
        partial += __shfl_xor(partial, off, 32);

    if (lane == 0)
        attn[(size_t)b * S_LEN + s] = partial;
}

// ---------------------------------------------------------------------------
// Kernel 3: softmax over s (dim=1) per batch b.  out[b,0,s].
// Note: the hidden.b_attn term is constant in s -> softmax shift-invariant,
// so the bias is dropped exactly.
// ---------------------------------------------------------------------------
__global__ __launch_bounds__(256)
void softmax_kernel(const float* __restrict__ attn,
                    float* __restrict__ out) {
    __shared__ float red[256];
    const int b = blockIdx.x;
    const int t = threadIdx.x;

    float x[8];
    float m = -INFINITY;
    #pragma unroll
    for (int j = 0; j < 8; ++j) {
        x[j] = attn[(size_t)b * S_LEN + t + j * 256];
        m = fmaxf(m, x[j]);
    }
    red[t] = m;
    __syncthreads();
    for (int stride = 128; stride > 0; stride >>= 1) {
        if (t < stride) red[t] = fmaxf(red[t], red[t + stride]);
        __syncthreads();
    }
    m = red[0];
    __syncthreads();

    float sum = 0.f;
    #pragma unroll
    for (int j = 0; j < 8; ++j) {
        x[j] = __expf(x[j] - m);
        sum += x[j];
    }
    red[t] = sum;
    __syncthreads();
    for (int stride = 128; stride > 0; stride >>= 1) {
        if (t < stride) red[t] += red[t + stride];
        __syncthreads();
    }
    const float inv = 1.0f / red[0];
    #pragma unroll
    for (int j = 0; j < 8; ++j)
        out[(size_t)b * S_LEN + t + j * 256] = x[j] * inv;
}

// ---------------------------------------------------------------------------
extern "C" void kernel_launch(void* const* d_in, const int* in_sizes, int n_in,
                              void* d_out, int out_size, void* d_ws, size_t ws_size,
                              hipStream_t stream) {
    (void)in_sizes; (void)n_in; (void)out_size; (void)ws_size;

    const float* hidden = (const float*)d_in[0];   // [1,B,H]
    const float* enc    = (const float*)d_in[1];   // [S,B,H]
    const float* W      = (const float*)d_in[2];   // [H,H]
    // d_in[3] = b_attn: mathematically cancels under softmax (shift-invariance)

    float* out  = (float*)d_out;                   // [B,1,S]
    float* v    = (float*)d_ws;                    // [B,H]   = 128 KB
    float* attn = v + (size_t)B_LEN * H_LEN;       // [B,S]   = 256 KB

    // 1) v = hidden @ W via fp32 WMMA (64 h-tiles x 2 b-tiles of 16x16)
    proj_wmma<<<dim3(H_LEN / 16, B_LEN / 16), 32, 0, stream>>>(hidden, W, v);

    // 2) attn energies: stream enc (HBM-bound, ~256 MB)
    energies_kernel<<<dim3(S_LEN / 8, B_LEN), 256, 0, stream>>>(enc, v, attn);

    // 3) softmax over sequence dim
    softmax_kernel<<<B_LEN, 256, 0, stream>>>(attn, out);
}